// P_update_40647570489989
// MI455X (gfx1250) — compile-verified
//
#include <hip/hip_runtime.h>
#include <hip/hip_bf16.h>

// ---------------------------------------------------------------------------
// P=2 norm-message scatter-mean over a 100K-node / 3.2M-edge graph, d=16.
// Memory-bound: 51.2 MB edge stream from HBM (~2.2us floor at 23.3 TB/s);
// x (6.4 MB) + accumulator (6.8 MB) are L2-resident (192 MB L2).
// Hot path = 17 fp32 atomics/edge into L2. CDNA5 feature used: async
// global->LDS double-buffering of the edge-index stream (ASYNCcnt).
// Edge kernel placed first in the TU so the disasm snippet shows its body.
// ---------------------------------------------------------------------------

#ifndef __has_builtin
#define __has_builtin(x) 0
#endif

#if defined(__gfx1250__) && \
    __has_builtin(__builtin_amdgcn_global_load_async_to_lds_b64) && \
    __has_builtin(__builtin_amdgcn_s_wait_asynccnt)
#define USE_ASYNC 1
#else
#define USE_ASYNC 0
#endif

#define TPB 256
#define WPB (TPB / 32)
#define EDGE_BLOCKS 2048

__device__ __forceinline__ void atomic_fadd(float* addr, float v) {
  __hip_atomic_fetch_add(addr, v, __ATOMIC_RELAXED, __HIP_MEMORY_SCOPE_AGENT);
}

#if USE_ASYNC
typedef int v2i __attribute__((vector_size(8)));

__device__ __forceinline__ void async_copy8(const void* g, void* l) {
  __builtin_amdgcn_global_load_async_to_lds_b64(
      (__attribute__((address_space(1))) v2i*)g,
      (__attribute__((address_space(3))) v2i*)l,
      /*offset=*/0, /*cpol=*/0);
}
#endif

__device__ __forceinline__ void process_edge(const float* __restrict__ x,
                                             float* __restrict__ agg,
                                             float* __restrict__ cnt,
                                             int src, int dst) {
  const float4* xs = (const float4*)(x + (size_t)src * 16);
  const float4* xd = (const float4*)(x + (size_t)dst * 16);
  float4 a0 = xs[0], a1 = xs[1], a2 = xs[2], a3 = xs[3];
  float4 b0 = xd[0], b1 = xd[1], b2 = xd[2], b3 = xd[3];

  float d[16];
  d[0]  = a0.x - b0.x; d[1]  = a0.y - b0.y; d[2]  = a0.z - b0.z; d[3]  = a0.w - b0.w;
  d[4]  = a1.x - b1.x; d[5]  = a1.y - b1.y; d[6]  = a1.z - b1.z; d[7]  = a1.w - b1.w;
  d[8]  = a2.x - b2.x; d[9]  = a2.y - b2.y; d[10] = a2.z - b2.z; d[11] = a2.w - b2.w;
  d[12] = a3.x - b3.x; d[13] = a3.y - b3.y; d[14] = a3.z - b3.z; d[15] = a3.w - b3.w;

  float sq = 0.0f;
#pragma unroll
  for (int i = 0; i < 16; ++i) sq = fmaf(d[i], d[i], sq);

  // Reference: msg = -diff / sq  (P=2). Self-loop => -0/0 = NaN, same as JAX.
  float s = -1.0f / sq;

  float* row = agg + (size_t)dst * 16;
#pragma unroll
  for (int i = 0; i < 16; ++i) atomic_fadd(row + i, d[i] * s);
  atomic_fadd(cnt + dst, 1.0f);
}

__global__ __launch_bounds__(TPB) void pnorm_edge_kernel(
    const float* __restrict__ x,
    const long long* __restrict__ ei,   // [2, E]: row 0 = src, row 1 = dst
    float* __restrict__ agg,
    float* __restrict__ cnt,
    long long E) {
  const int lane = (int)threadIdx.x & 31;
  const int wv   = (int)threadIdx.x >> 5;

  const long long wavesTotal  = (long long)gridDim.x * WPB;
  const long long waveId      = (long long)blockIdx.x * WPB + wv;
  const long long chunkStride = wavesTotal * 32;
  long long chunk = waveId * 32;

#if USE_ASYNC
  // Per-wave double-buffered staging of the sequential edge-index stream.
  // Each lane async-copies 8B (one int64 index); 2 async ops per stage issue.
  __shared__ long long sbuf[2][WPB][32];
  __shared__ long long dbuf[2][WPB][32];

  int stage = 0;
  if (chunk < E) {
    async_copy8(ei + chunk + lane,     &sbuf[0][wv][lane]);
    async_copy8(ei + E + chunk + lane, &dbuf[0][wv][lane]);
  }
  for (; chunk < E; chunk += chunkStride) {
    const long long next = chunk + chunkStride;
    if (next < E) {
      async_copy8(ei + next + lane,     &sbuf[stage ^ 1][wv][lane]);
      async_copy8(ei + E + next + lane, &dbuf[stage ^ 1][wv][lane]);
      __builtin_amdgcn_s_wait_asynccnt(2);  // current stage done (in-order)
    } else {
      __builtin_amdgcn_s_wait_asynccnt(0);
    }
    const int src = (int)sbuf[stage][wv][lane];
    const int dst = (int)dbuf[stage][wv][lane];
    stage ^= 1;
    process_edge(x, agg, cnt, src, dst);
  }
#else
  for (; chunk < E; chunk += chunkStride) {
    const int src = (int)ei[chunk + lane];
    const int dst = (int)ei[E + chunk + lane];
    process_edge(x, agg, cnt, src, dst);
  }
#endif
}

__global__ void zero_ws_kernel(float* __restrict__ p, int n) {
  int i = blockIdx.x * blockDim.x + threadIdx.x;
  if (i < n) p[i] = 0.0f;
}

__global__ void finalize_kernel(const float* __restrict__ agg,
                                const float* __restrict__ cnt,
                                const float* __restrict__ ratio,
                                float* __restrict__ out, int n16) {
  int i = blockIdx.x * blockDim.x + threadIdx.x;
  if (i < n16) {
    const int node = i >> 4;
    float c = cnt[node];
    c = c > 1.0f ? c : 1.0f;          // max(cnt, 1): isolated nodes -> 0
    out[i] = ratio[0] * (agg[i] / c);
  }
}

extern "C" void kernel_launch(void* const* d_in, const int* in_sizes, int n_in,
                              void* d_out, int out_size, void* d_ws, size_t ws_size,
                              hipStream_t stream) {
  const float*     x     = (const float*)d_in[0];
  const float*     ratio = (const float*)d_in[1];
  const long long* ei    = (const long long*)d_in[2];

  const int       N   = in_sizes[0] / 16;
  const long long E   = (long long)in_sizes[2] / 2;
  const int       n16 = N * 16;

  float* agg = (float*)d_ws;        // N*16 floats
  float* cnt = agg + n16;           // N floats

  const int nzero = n16 + N;
  zero_ws_kernel<<<(nzero + TPB - 1) / TPB, TPB, 0, stream>>>(agg, nzero);

  pnorm_edge_kernel<<<EDGE_BLOCKS, TPB, 0, stream>>>(x, ei, agg, cnt, E);

  finalize_kernel<<<(n16 + TPB - 1) / TPB, TPB, 0, stream>>>(
      agg, cnt, ratio, (float*)d_out, n16);
}